// Amatrix_7352984011045
// MI455X (gfx1250) — compile-verified
//
#include <hip/hip_runtime.h>
#include <cstdint>
#include <cstddef>

#define C_DIM 128
#define H_DIM 2
#define L_DIM 3
#define NEG_SLOPE 0.2f
#define NWAVES 4   // waves per block in the WMMA logits kernel (LDS: 4*8KB + 8KB)

typedef __attribute__((ext_vector_type(2))) float v2f;
typedef __attribute__((ext_vector_type(8))) float v8f;
typedef __attribute__((ext_vector_type(4))) int   v4i;

__device__ __forceinline__ float lrelu(float v) {
  return v >= 0.f ? v : NEG_SLOPE * v;
}

// float atomic max via sign-aware integer atomics (works with -inf init)
__device__ __forceinline__ void atomicMaxF32(float* addr, float v) {
  if (!(v < 0.f)) {
    atomicMax((int*)addr, __float_as_int(v));
  } else {
    atomicMin((unsigned int*)addr, (unsigned int)__float_as_int(v));
  }
}

__device__ __forceinline__ void atomicAddF32(float* addr, float v) {
  __hip_atomic_fetch_add(addr, v, __ATOMIC_RELAXED, __HIP_MEMORY_SCOPE_AGENT);
}

// ---------------- init kernels ----------------
__global__ void k_init_base(const float* __restrict__ x, float* __restrict__ cur,
                            float* __restrict__ out, int nc) {
  int i = blockIdx.x * blockDim.x + threadIdx.x;
  if (i >= nc) return;
  float v = x[i];
  cur[i] = v;
  out[i] = v;  // accumulates x0 + x1 + x2 + x3
}

__global__ void k_init_layer(float* __restrict__ nxt, float* __restrict__ mbuf,
                             float* __restrict__ dn, int nc, int nh) {
  int i = blockIdx.x * blockDim.x + threadIdx.x;
  if (i >= nc) return;
  nxt[i] = 0.f;
  if (i < nh) {
    mbuf[i] = __int_as_float(0xff800000);  // -inf
    dn[i] = 0.f;
  }
}

// ---------------- edge logits via WMMA + segment max ----------------
__global__ __launch_bounds__(32 * NWAVES) void k_logits(
    const float* __restrict__ cur, const int* __restrict__ srcI,
    const int* __restrict__ dstI, const float* __restrict__ attL,
    float* __restrict__ eb, float* __restrict__ mbuf, int E, int numTiles) {
  __shared__ float sS[NWAVES][16][C_DIM];  // per-wave: 16 edges x 128 ch of s
  __shared__ float attS[16][C_DIM];        // rows 0..1 = att; rows 2..15 = 0

  const int t = threadIdx.x;
  const int wave = t >> 5;
  const int lane = t & 31;

  // ---- zero pad rows of B so the WMMA B-load needs no predication ----
  {
    float4* attV = (float4*)&attS[0][0];  // 512 float4s total
#pragma unroll
    for (int i = (H_DIM * C_DIM) / 4 + t; i < 16 * C_DIM / 4; i += 32 * NWAVES) {
      attV[i] = make_float4(0.f, 0.f, 0.f, 0.f);
    }
  }

  // ---- stage att (256 floats) into LDS; CDNA5 async-to-LDS path ----
#if __has_builtin(__builtin_amdgcn_global_load_async_to_lds_b128)
  if (t < (H_DIM * C_DIM) / 4) {
    __builtin_amdgcn_global_load_async_to_lds_b128(
        (__attribute__((address_space(1))) v4i*)(attL + t * 4),
        (__attribute__((address_space(3))) v4i*)(&attS[0][0] + t * 4),
        /*offset=*/0, /*cpol=*/0);
  }
#if __has_builtin(__builtin_amdgcn_s_wait_asynccnt)
  __builtin_amdgcn_s_wait_asynccnt(0);
#else
  asm volatile("s_wait_asynccnt 0" ::: "memory");
#endif
#else
  if (t < (H_DIM * C_DIM) / 4) {
    ((float4*)&attS[0][0])[t] = ((const float4*)attL)[t];
  }
#endif

  const int tile = blockIdx.x * NWAVES + wave;
  const bool valid = tile < numTiles;

  // ---- build s = leaky_relu(x[dst]+x[src]) for 16 edges into LDS ----
  if (valid) {
    int e = tile * 16 + (lane >> 1);
    if (e >= E) e = E - 1;  // clamp: duplicate row in LDS, never extracted
    const int si = srcI[e], di = dstI[e];
    const float4* xs = (const float4*)(cur + (size_t)si * C_DIM);
    const float4* xd = (const float4*)(cur + (size_t)di * C_DIM);
    float4* srow = (float4*)(&sS[wave][lane >> 1][0]);
    const int cb = (lane & 1) * 16;  // which 64-channel half (in float4 units)
#pragma unroll
    for (int j = 0; j < 16; ++j) {
      float4 a = xs[cb + j];
      float4 b = xd[cb + j];
      float4 v;
      v.x = lrelu(a.x + b.x);
      v.y = lrelu(a.y + b.y);
      v.z = lrelu(a.z + b.z);
      v.w = lrelu(a.w + b.w);
      srow[cb + j] = v;
    }
  }
  __syncthreads();
  if (!valid) return;

  const int n16 = lane & 15;    // A: edge row (M);  B/D: column (N)
  const int khalf = lane >> 4;  // selects K pair {0,1} vs {2,3} inside each K4 block

#if __has_builtin(__builtin_amdgcn_wmma_f32_16x16x4_f32)
  // D[M=edge][N] = sum_K A[M][K]*B[K][N], B[K][n] = att[n][K] (zero for n>=H)
  const float* sa = &sS[wave][n16][2 * khalf];
  const float* bb = &attS[n16][2 * khalf];
  v8f acc = {0.f, 0.f, 0.f, 0.f, 0.f, 0.f, 0.f, 0.f};
#pragma unroll
  for (int k = 0; k < C_DIM / 4; ++k) {
    v2f a = *(const v2f*)(sa + 4 * k);
    v2f b = *(const v2f*)(bb + 4 * k);
    acc = __builtin_amdgcn_wmma_f32_16x16x4_f32(
        /*neg_a=*/false, a, /*neg_b=*/false, b,
        /*c_mod=*/(short)0, acc, /*reuse_a=*/false, /*reuse_b=*/false);
  }
  // D layout: lane L holds D[M = v + 8*(L>>4)][N = L&15] in acc[v]
  if (n16 < H_DIM) {
    const int ebase = tile * 16 + 8 * khalf;
#pragma unroll
    for (int v = 0; v < 8; ++v) {
      const int e = ebase + v;
      if (e < E) {
        float val = acc[v];
        eb[(size_t)e * H_DIM + n16] = val;
        atomicMaxF32(&mbuf[(size_t)dstI[e] * H_DIM + n16], val);
      }
    }
  }
#else
  // VALU fallback: lane -> (edge = lane>>1, head = lane&1)
  (void)khalf;
  const int eloc = lane >> 1;
  const int h = lane & 1;
  const int e = tile * 16 + eloc;
  if (e < E) {
    float acc0 = 0.f;
#pragma unroll 16
    for (int c = 0; c < C_DIM; ++c) acc0 += sS[wave][eloc][c] * attS[h][c];
    eb[(size_t)e * H_DIM + h] = acc0;
    atomicMaxF32(&mbuf[(size_t)dstI[e] * H_DIM + h], acc0);
  }
  (void)n16;
#endif
}

// ---------------- exp + segment sum ----------------
__global__ void k_exp(const int* __restrict__ dstI, float* __restrict__ eb,
                      const float* __restrict__ mbuf, float* __restrict__ dn,
                      int EH) {
  int i = blockIdx.x * blockDim.x + threadIdx.x;
  if (i >= EH) return;
  int e = i >> 1, h = i & 1;
  int d = dstI[e];
  float v = expf(eb[i] - mbuf[d * H_DIM + h]);
  eb[i] = v;
  atomicAddF32(&dn[d * H_DIM + h], v);
}

// ---------------- alpha = e/denom ; emit alphas; amean for scatter ----------------
__global__ void k_alpha(const int* __restrict__ dstI, float* __restrict__ eb,
                        const float* __restrict__ dn, float* __restrict__ alphaOut,
                        int E) {
  int e = blockIdx.x * blockDim.x + threadIdx.x;
  if (e >= E) return;
  int d = dstI[e];
  float a0 = eb[2 * (size_t)e] / dn[d * 2];
  float a1 = eb[2 * (size_t)e + 1] / dn[d * 2 + 1];
  alphaOut[2 * (size_t)e] = a0;
  alphaOut[2 * (size_t)e + 1] = a1;
  eb[2 * (size_t)e] = 0.5f * (a0 + a1);  // mean over heads (x_j is head-indep)
}

// ---------------- scatter-aggregate: nxt[dst] += x[src] * amean ----------------
__global__ void k_scatter(const int* __restrict__ srcI, const int* __restrict__ dstI,
                          const float* __restrict__ cur, const float* __restrict__ eb,
                          float* __restrict__ nxt, int E) {
  int idx = blockIdx.x * blockDim.x + threadIdx.x;
  if (idx >= E * 4) return;
  int e = idx >> 2, q = idx & 3;  // 4 threads per edge, 32 channels each
  float am = eb[2 * (size_t)e];
  int si = srcI[e], di = dstI[e];
  const float4* xr = (const float4*)(cur + (size_t)si * C_DIM + q * 32);
  float* orow = nxt + (size_t)di * C_DIM + q * 32;
#pragma unroll
  for (int j = 0; j < 8; ++j) {
    float4 v = xr[j];
    atomicAddF32(&orow[4 * j + 0], v.x * am);
    atomicAddF32(&orow[4 * j + 1], v.y * am);
    atomicAddF32(&orow[4 * j + 2], v.z * am);
    atomicAddF32(&orow[4 * j + 3], v.w * am);
  }
}

// ---------------- cur = nxt + bias ; out += cur ----------------
__global__ void k_update(const float* __restrict__ nxt, const float* __restrict__ biasL,
                         float* __restrict__ cur, float* __restrict__ out, int nc) {
  int i = blockIdx.x * blockDim.x + threadIdx.x;
  if (i >= nc) return;
  float v = nxt[i] + biasL[i & (C_DIM - 1)];
  cur[i] = v;
  out[i] += v;
}

__global__ void k_final(float* __restrict__ out, int nc) {
  int i = blockIdx.x * blockDim.x + threadIdx.x;
  if (i >= nc) return;
  out[i] *= 1.0f / (float)(L_DIM + 1);
}

extern "C" void kernel_launch(void* const* d_in, const int* in_sizes, int n_in,
                              void* d_out, int out_size, void* d_ws, size_t ws_size,
                              hipStream_t stream) {
  const float* x = (const float*)d_in[0];
  const int* ei = (const int*)d_in[1];
  const float* att = (const float*)d_in[2];
  const float* bias = (const float*)d_in[3];

  const int N = in_sizes[0] / C_DIM;
  const int E = in_sizes[1] / 2;
  const int NC = N * C_DIM;
  const int NH = N * H_DIM;
  const int EH = E * H_DIM;

  float* out = (float*)d_out;             // [N, C]
  float* alphas = out + (size_t)NC;       // [L, E, H]

  float* cur = (float*)d_ws;              // [N, C]
  float* nxt = cur + (size_t)NC;          // [N, C]
  float* mbuf = nxt + (size_t)NC;         // [N, H] segment max
  float* dn = mbuf + (size_t)NH;          // [N, H] segment sum
  float* eb = dn + (size_t)NH;            // [E, H] logits -> exp -> amean

  const int* srcI = ei;
  const int* dstI = ei + E;

  const int B = 256;
  const int numTiles = (E + 15) / 16;

  k_init_base<<<(NC + B - 1) / B, B, 0, stream>>>(x, cur, out, NC);

  for (int l = 0; l < L_DIM; ++l) {
    k_init_layer<<<(NC + B - 1) / B, B, 0, stream>>>(nxt, mbuf, dn, NC, NH);
    k_logits<<<(numTiles + NWAVES - 1) / NWAVES, 32 * NWAVES, 0, stream>>>(
        cur, srcI, dstI, att + (size_t)l * H_DIM * C_DIM, eb, mbuf, E, numTiles);
    k_exp<<<(EH + B - 1) / B, B, 0, stream>>>(dstI, eb, mbuf, dn, EH);
    k_alpha<<<(E + B - 1) / B, B, 0, stream>>>(dstI, eb, dn,
                                               alphas + (size_t)l * EH, E);
    k_scatter<<<(E * 4 + B - 1) / B, B, 0, stream>>>(srcI, dstI, cur, eb, nxt, E);
    k_update<<<(NC + B - 1) / B, B, 0, stream>>>(nxt, bias + (size_t)l * C_DIM,
                                                 cur, out, NC);
  }
  k_final<<<(NC + B - 1) / B, B, 0, stream>>>(out, NC);
}